// UltraHighPrecisionNKATHamiltonian_16441134809564
// MI455X (gfx1250) — compile-verified
//
#include <hip/hip_runtime.h>
#include <cstdint>

// ---------------------------------------------------------------------------
// UltraHighPrecisionNKATHamiltonian for MI455X (gfx1250)
//
// H[b] = Cherm (constant 300x300 hermitized correction matrix) with the
// diagonal replaced by Re(w[b,i]) + Re(BASE[i,i]) + 1e-14 (purely real).
//
// Roofline: 368.6 MB of complex128 stores -> ~15.8 us at 23.3 TB/s. Compute is
// negligible, so the kernel is built around the CDNA5 DMA paths:
//   tensor_load_to_lds            (TDM: stage constant tile into LDS, TENSORcnt)
//   global_store_async_from_lds_b128 (broadcast LDS -> 16 batch slices, ASYNCcnt)
// No WMMA: CDNA5 has no f64 WMMA shape and this problem is 99.9% bandwidth.
// ---------------------------------------------------------------------------

#define DIM 300
#define ZETA_TERMS 100
#define THETA 1e-20
#define KAPPA 1e-12
#define RICCI_C (6.0 / (24.0 * 3.14159265358979323846 * 3.14159265358979323846))

#define ROWS_PER_BLK 4     // rows of Cherm staged in LDS per workgroup
#define BATCH_PER_BLK 16   // batch slices written per workgroup
#define ELEMS_PER_BLK (ROWS_PER_BLK * DIM)  // 1200 complex128 = 19200 B LDS
#define ROW_QWORDS (DIM * 2)                // 600 8-byte elements per row

typedef unsigned int v4u __attribute__((ext_vector_type(4)));
typedef int          v8i __attribute__((ext_vector_type(8)));
typedef int          v4i __attribute__((ext_vector_type(4)));

// first 72 primes (reference uses primes[i], i <= 68)
__constant__ int PRIMES[72] = {
    2,3,5,7,11,13,17,19,23,29,31,37,41,43,47,53,59,61,67,71,
    73,79,83,89,97,101,103,107,109,113,127,131,137,139,149,151,157,163,167,173,
    179,181,191,193,197,199,211,223,227,229,233,239,241,251,257,263,269,271,277,281,
    283,293,307,311,313,317,331,337,347,349,353,359};

__device__ __forceinline__ double kappa_b(int i) {
    double n = (double)(i + 1);
    return KAPPA * n * log(n + 1.0) / (n + 1.0);
}

// element (i,j) of the raw (pre-hermitization) correction matrix C
__device__ void base_elem(int i, int j, double& re, double& im) {
    re = 0.0; im = 0.0;
    // theta (noncommutative) term
    if (i < j) {
        if (i < 50 && j < i + 20)
            im += THETA * (log((double)PRIMES[i]) - log((double)PRIMES[j]));
    } else if (j < i) {
        if (j < 50 && i < j + 20)   // C[j0][i0] -= conj(c) with c purely imaginary
            im += THETA * (log((double)PRIMES[j]) - log((double)PRIMES[i]));
    }
    // kappa (quantum field) corrections
    if (i == j) {
        if (i < 30) re += kappa_b(i);
        if (i < 40) { double n = (double)(i + 1); re += RICCI_C / (n * n); }
    }
    if (j == i + 1 && i < 30) re += 0.1  * kappa_b(i);
    if (i == j + 1 && j < 30) re += 0.1  * kappa_b(j);
    if (j == i + 2 && i < 30) re += 0.01 * kappa_b(i);
    if (i == j + 2 && j < 30) re += 0.01 * kappa_b(j);
    // spin connection: only gamma0[0,0]=1 contributes -> pairs (0,j), j=1..3
    if (i == 0 && j >= 1 && j <= 3) re += -0.01 * (double)j;
    if (j == 0 && i >= 1 && i <= 3) re += -0.01 * (double)i;
}

// Cherm = 0.5 * (C + C^H), stored interleaved {re,im}
__global__ __launch_bounds__(256) void build_cherm(double* __restrict__ cherm) {
    int idx = blockIdx.x * blockDim.x + threadIdx.x;
    if (idx >= DIM * DIM) return;
    int i = idx / DIM, j = idx % DIM;
    double ar, ai, br, bi;
    base_elem(i, j, ar, ai);
    base_elem(j, i, br, bi);
    cherm[2 * idx]     = 0.5 * (ar + br);
    cherm[2 * idx + 1] = 0.5 * (ai - bi);
}

// per-batch diagonal: Re(w) + Re(BASE[k,k]) + 1e-14
__global__ __launch_bounds__(128) void compute_diag(const double* __restrict__ s_real,
                                                    const double* __restrict__ s_imag,
                                                    double* __restrict__ dval) {
    const int b = blockIdx.x;
    const int t = threadIdx.x;
    const double sr = s_real[b], si = s_imag[b];

    __shared__ double red[2 * 128];
    double zr = 0.0, zi = 0.0;
    for (int n = t + 1; n <= ZETA_TERMS; n += 128) {
        double ln = log((double)n);
        double mag = exp(-sr * ln);
        double sn, cs; sincos(si * ln, &sn, &cs);
        zr += mag * cs; zi -= mag * sn;
    }
    red[t] = zr; red[128 + t] = zi;
    __syncthreads();
    for (int s = 64; s > 0; s >>= 1) {
        if (t < s) { red[t] += red[t + s]; red[128 + t] += red[128 + t + s]; }
        __syncthreads();
    }
    zr = red[0]; zi = red[128];

    const double az = sqrt(zr * zr + zi * zi);
    const bool big = az > 1e-15;
    const double pr = big ? zr / az : 0.0;
    const double pi = big ? zi / az : 0.0;

    for (int k = t; k < DIM; k += 128) {
        double ln = log((double)(k + 1));
        double mag = exp(-sr * ln);
        double sn, cs; sincos(si * ln, &sn, &cs);
        double br = mag * cs, bi = -mag * sn;
        double wr, wi;
        if (big) { wr = br * pr - bi * pi; wi = br * pi + bi * pr; }
        else     { wr = 1e-15; wi = 0.0; }
        double aw = sqrt(wr * wr + wi * wi);      // aw sampled BEFORE clamps (per ref)
        if      (aw < 1e-50) wr = 1e-50;
        else if (aw > 1e+50) wr = 1e+50;
        double d = wr + 1e-14;                    // Re(w) + 1e-14
        if (k < 30) d += kappa_b(k);              // Re(BASE[k,k]) pieces
        if (k < 40) { double n = (double)(k + 1); d += RICCI_C / (n * n); }
        dval[(size_t)b * DIM + k] = d;
    }
}

// ---- Tensor DMA descriptor builders (CDNA5 D#, CDNA5 ISA) ----
// Group0: [1:0]=count(1), [63:32]=lds_addr, [120:64]=global_addr, [127:126]=type(2)
__device__ __forceinline__ v4u tdm_group0(unsigned long long gaddr, unsigned lds_addr) {
    v4u g;
    g.x = 1u;                                       // count=1 valid, no gather
    g.y = lds_addr;                                 // LDS byte address
    g.z = (unsigned)(gaddr & 0xFFFFFFFFu);          // global_addr[31:0]
    g.w = (unsigned)((gaddr >> 32) & 0x1FFFFFFu)    // global_addr[56:32]
        | (2u << 30);                               // type = 2 ("image")
    return g;
}
// Group1: data_size[17:16]=3(8B), tensor_dim0[79:48], tensor_dim1[111:80],
//         tile_dim0[127:112], tile_dim1[143:128], tensor_dim0_stride[207:160]
__device__ __forceinline__ v8i tdm_group1_2d(unsigned td0, unsigned td1,
                                             unsigned tile0, unsigned tile1,
                                             unsigned long long stride0) {
    v8i g;
    g[0] = (int)(3u << 16);                                         // data_size=8B
    g[1] = (int)((td0 & 0xFFFFu) << 16);
    g[2] = (int)(((td0 >> 16) & 0xFFFFu) | ((td1 & 0xFFFFu) << 16));
    g[3] = (int)(((td1 >> 16) & 0xFFFFu) | ((tile0 & 0xFFFFu) << 16));
    g[4] = (int)(tile1 & 0xFFFFu);                                  // tile_dim2 = 0
    g[5] = (int)(unsigned)(stride0 & 0xFFFFFFFFu);
    g[6] = (int)(unsigned)((stride0 >> 32) & 0xFFFFu);              // stride1 = 0
    g[7] = 0;
    return g;
}

// Bulk broadcast: TDM-load ROWS_PER_BLK rows of Cherm into LDS (one descriptor),
// then replicate to BATCH_PER_BLK batch slices via async store-from-LDS.
// NOTE: single __shared__ array in this kernel -> it sits at LDS offset 0,
// which is what the DMA instructions' LDS addresses encode.
__global__ __launch_bounds__(256) void broadcast(const double* __restrict__ cherm,
                                                 double* __restrict__ out, int nbatch) {
    __shared__ double sh[ELEMS_PER_BLK * 2];  // 19200 bytes at LDS offset 0
    // keep the LDS allocation alive (array is only touched by the DMA engine)
    unsigned sh_keep = (unsigned)(size_t)&sh[0];
    asm volatile("" :: "v"(sh_keep));

    const int tid  = threadIdx.x;
    const int row0 = blockIdx.x * ROWS_PER_BLK;      // 75 chunks * 4 = 300 rows
    const int b0   = blockIdx.y * BATCH_PER_BLK;

    // ---- TDM: one descriptor stages the whole 4x4800B tile (TENSORcnt) ----
    const unsigned long long gbase =
        (unsigned long long)(cherm + (size_t)row0 * DIM * 2);
    if (tid < 32) {                  // exactly one wave issues the DMA
        v4u g0 = tdm_group0(gbase, 0u);
        v8i g1 = tdm_group1_2d(/*tensor_dim0=*/ROW_QWORDS, /*tensor_dim1=*/ROWS_PER_BLK,
                               /*tile_dim0=*/ROW_QWORDS,   /*tile_dim1=*/ROWS_PER_BLK,
                               /*stride0=*/ROW_QWORDS);
        v4i z4 = {0, 0, 0, 0};       // groups 2/3 unused (2D tensor)
        v8i z8 = {0, 0, 0, 0, 0, 0, 0, 0};
        // 6-arg variant on this toolchain: (g0, g1, g2, g3, g_extra, cpol)
        __builtin_amdgcn_tensor_load_to_lds(g0, g1, z4, z4, z8, 0);
    }
    __builtin_amdgcn_s_wait_tensorcnt(0);
    __syncthreads();   // issuing wave drained TENSORcnt -> LDS tile valid for all

    // ---- async store: LDS -> global, replicated across batches (ASYNCcnt) ----
    const int bend = (b0 + BATCH_PER_BLK < nbatch) ? b0 + BATCH_PER_BLK : nbatch;
    for (int b = b0; b < bend; ++b) {
        const unsigned long long obase =
            (unsigned long long)(out + ((size_t)b * DIM + row0) * DIM * 2);
        for (int e = tid; e < ELEMS_PER_BLK; e += 256) {
            unsigned off = (unsigned)(e * 16);
            asm volatile("global_store_async_from_lds_b128 %0, %1, %2"
                         :: "v"(off), "v"(off), "s"(obase) : "memory");
        }
    }
    __builtin_amdgcn_s_wait_asynccnt(0);
}

// overwrite the diagonal (256*300 elements, 0.3% of output traffic)
__global__ __launch_bounds__(320) void fix_diag(const double* __restrict__ dval,
                                                double* __restrict__ out) {
    const int b = blockIdx.x, i = threadIdx.x;
    if (i < DIM) {
        size_t idx = (((size_t)b * DIM + i) * DIM + i) * 2;
        out[idx]     = dval[(size_t)b * DIM + i];
        out[idx + 1] = 0.0;   // hermitization zeroes Im on the diagonal
    }
}

extern "C" void kernel_launch(void* const* d_in, const int* in_sizes, int n_in,
                              void* d_out, int out_size, void* d_ws, size_t ws_size,
                              hipStream_t stream) {
    const double* s_real = (const double*)d_in[0];   // float64
    const double* s_imag = (const double*)d_in[1];   // float64
    double* out = (double*)d_out;                    // complex128 -> interleaved f64
    const int nbatch = in_sizes[0];                  // 256

    // workspace: [Cherm: 300*300 complex128 = 1.44 MB][dval: B*300 f64]
    double* cherm = (double*)d_ws;
    double* dval  = (double*)((char*)d_ws + (size_t)DIM * DIM * 2 * sizeof(double));

    build_cherm<<<(DIM * DIM + 255) / 256, 256, 0, stream>>>(cherm);
    compute_diag<<<nbatch, 128, 0, stream>>>(s_real, s_imag, dval);
    dim3 grid(DIM / ROWS_PER_BLK, (nbatch + BATCH_PER_BLK - 1) / BATCH_PER_BLK);
    broadcast<<<grid, 256, 0, stream>>>(cherm, out, nbatch);
    fix_diag<<<nbatch, 320, 0, stream>>>(dval, out);
}